// ProposalCreator_12687333392928
// MI455X (gfx1250) — compile-verified
//
#include <hip/hip_runtime.h>
#include <stdint.h>

typedef unsigned int u32;

#define N_ELE    1048576
#define NPRE     6000
#define NPOST    300
#define CAND_CAP 8192
#define MWORDS   188           // ceil(6000/32)
#define TPB      256

// ---------------- workspace layout (bytes) ----------------
#define OFF_ROI        0u                     // float4[N]        16,777,216
#define OFF_KEY        16777216u              // u32[N]            4,194,304
#define OFF_HIST1      20971520u              // u32[4096]
#define OFF_HIST2      20987904u              // u32[4096]
#define OFF_HIST3      21004288u              // u32[256]
#define OFF_SEL        21005312u              // u32[8]
#define OFF_VWORDS     21005568u              // u32[188]
#define OFF_CKEY       21006336u              // u32[8192]
#define OFF_CIDX       21039104u              // u32[8192]
#define OFF_SROI       21071872u              // float4[NPRE]      96,000
#define OFF_MASK       21167872u              // u32[NPRE*188]   4,512,000
// total ~24.5 MB

// ---------------- CDNA5 async global->LDS helpers ----------------
__device__ __forceinline__ void async_ld_b32(u32 lds_byte_off, const void* gptr) {
    asm volatile("global_load_async_to_lds_b32 %0, %1, off"
                 :: "v"(lds_byte_off), "v"(gptr) : "memory");
}
__device__ __forceinline__ void async_ld_b128(u32 lds_byte_off, const void* gptr) {
    asm volatile("global_load_async_to_lds_b128 %0, %1, off"
                 :: "v"(lds_byte_off), "v"(gptr) : "memory");
}
__device__ __forceinline__ void wait_async0() {
    asm volatile("s_wait_asynccnt 0" ::: "memory");
}

// ---------------- kernels ----------------
__global__ void zero_u32(u32* p, int n) {
    int i = blockIdx.x * blockDim.x + threadIdx.x;
    if (i < n) p[i] = 0u;
}

// decode boxes, build monotone sort key, level-1 histogram (top 12 bits)
__global__ void decode_hist1(const float4* __restrict__ loc,
                             const float*  __restrict__ score,
                             const float4* __restrict__ anchor,
                             const int* __restrict__ imgh, const int* __restrict__ imgw,
                             float4* __restrict__ roi, u32* __restrict__ key,
                             u32* __restrict__ hist1) {
    int i = blockIdx.x * blockDim.x + threadIdx.x;
    if (i >= N_ELE) return;
    float H = (float)imgh[0], W = (float)imgw[0];
    float4 a = anchor[i], l = loc[i];
    float h  = a.z - a.x, w = a.w - a.y;
    float cy = a.x + 0.5f * h, cx = a.y + 0.5f * w;
    float ncy = fmaf(l.x, h, cy), ncx = fmaf(l.y, w, cx);
    float nh = expf(l.z) * h, nw = expf(l.w) * w;
    float y1 = fminf(fmaxf(ncy - 0.5f * nh, 0.f), H);
    float x1 = fminf(fmaxf(ncx - 0.5f * nw, 0.f), W);
    float y2 = fminf(fmaxf(ncy + 0.5f * nh, 0.f), H);
    float x2 = fminf(fmaxf(ncx + 0.5f * nw, 0.f), W);
    bool valid = ((y2 - y1) >= 16.0f) && ((x2 - x1) >= 16.0f);
    u32 k = valid ? (__float_as_uint(score[i]) ^ 0x80000000u) : 0u;  // scores >= 0
    roi[i] = make_float4(y1, x1, y2, x2);
    key[i] = k;
    atomicAdd(&hist1[k >> 20], 1u);
}

__global__ void hist2_kernel(const u32* __restrict__ key, const u32* __restrict__ sel,
                             u32* __restrict__ hist2) {
    int i = blockIdx.x * blockDim.x + threadIdx.x;
    if (i >= N_ELE) return;
    u32 k = key[i];
    if ((k >> 20) == sel[0]) atomicAdd(&hist2[(k >> 8) & 0xFFFu], 1u);
}

__global__ void hist3_kernel(const u32* __restrict__ key, const u32* __restrict__ sel,
                             u32* __restrict__ hist3) {
    int i = blockIdx.x * blockDim.x + threadIdx.x;
    if (i >= N_ELE) return;
    u32 k = key[i];
    if ((k >> 8) == ((sel[0] << 12) | sel[2])) atomicAdd(&hist3[k & 0xFFu], 1u);
}

// scan a histogram from the top to locate bin containing the NPRE-th largest key
__global__ void scan_kernel(const u32* __restrict__ hist, int nbins,
                            u32* __restrict__ sel, int phase) {
    if (threadIdx.x != 0 || blockIdx.x != 0) return;
    u32 target = (phase == 0) ? (u32)NPRE
               : (phase == 1) ? ((u32)NPRE - sel[1])
                              : ((u32)NPRE - sel[3]);
    u32 cum = 0, above = 0; int found = -1;
    for (int b = nbins - 1; b >= 0; --b) {
        u32 c = hist[b];
        if (cum + c >= target) { found = b; above = cum; break; }
        cum += c;
    }
    if (found < 0) { found = 0; above = cum; }   // degenerate safety
    if (phase == 0)      { sel[0] = (u32)found; sel[1] = above; }
    else if (phase == 1) { sel[2] = (u32)found; sel[3] = sel[1] + above; }
    else                 { sel[4] = (sel[0] << 20) | (sel[2] << 8) | (u32)found; }
}

// gather all elements with key >= threshold (unordered, capped)
__global__ void gather_kernel(const u32* __restrict__ key, u32* __restrict__ sel,
                              u32* __restrict__ ckey, u32* __restrict__ cidx) {
    int i = blockIdx.x * blockDim.x + threadIdx.x;
    if (i >= N_ELE) return;
    u32 k = key[i];
    if (k >= sel[4]) {
        u32 p = atomicAdd(&sel[5], 1u);
        if (p < CAND_CAP) { ckey[p] = k; cidx[p] = (u32)i; }
    }
}

// exact stable rank (== argsort(-score) order, ties by index) + scatter sorted rois
__global__ void rank_kernel(const u32* __restrict__ ckey, const u32* __restrict__ cidx,
                            const u32* __restrict__ sel, const float4* __restrict__ roi,
                            float4* __restrict__ sroi, u32* __restrict__ vwords) {
    __shared__ u32 skey[CAND_CAP];
    __shared__ u32 sidx[CAND_CAP];
    int t = threadIdx.x;
    u32 M = sel[5]; if (M > CAND_CAP) M = CAND_CAP;
    for (int e = t; e < (int)M; e += TPB) {
        async_ld_b32((u32)(uintptr_t)&skey[e], &ckey[e]);
        async_ld_b32((u32)(uintptr_t)&sidx[e], &cidx[e]);
    }
    wait_async0();
    __syncthreads();
    int c = blockIdx.x * TPB + t;
    if (c < (int)M) {
        u32 mk = skey[c], mi = sidx[c], rank = 0;
        for (u32 e = 0; e < M; ++e) {
            u32 k = skey[e];
            rank += (k > mk) ? 1u : 0u;
            rank += (k == mk && sidx[e] < mi) ? 1u : 0u;
        }
        if (rank < (u32)NPRE) {
            sroi[rank] = roi[mi];
            if (mk & 0x80000000u)   // valid flag (non -inf score)
                atomicOr(&vwords[rank >> 5], 1u << (rank & 31));
        }
    }
}

// 6000x6000 IoU>thresh bitmask. 8 waves/block: 256 rows x 32 cols per block.
// Wave 0 stages column boxes via async global->LDS, precomputes col areas.
// Blocks whose column word lies strictly below the block's row range are skipped
// (those words are never read by the sequential resolver).
__global__ void __launch_bounds__(256)
nms_mask(const float4* __restrict__ sroi, u32* __restrict__ mask) {
    __shared__ float4 cbox[32];
    __shared__ float  carea[32];
    int lane   = threadIdx.x;     // 0..31
    int waveid = threadIdx.y;     // 0..7
    int rowBase = blockIdx.y * 256;
    if ((int)blockIdx.x < (rowBase >> 5)) return;   // uniform per block: safe w.r.t. barrier
    if (waveid == 0) {
        int cc = blockIdx.x * 32 + lane; if (cc >= NPRE) cc = NPRE - 1;
        async_ld_b128((u32)(uintptr_t)&cbox[lane], &sroi[cc]);
        wait_async0();                               // single wave's own asynccnt
        float4 q = cbox[lane];
        carea[lane] = (q.z - q.x) * (q.w - q.y);
    }
    __syncthreads();
    int r = rowBase + waveid * 32 + lane;
    if (r >= NPRE) return;
    float4 b = sroi[r];
    float ar = (b.z - b.x) * (b.w - b.y);
    u32 bits = 0;
    #pragma unroll 4
    for (int j = 0; j < 32; ++j) {
        float4 q = cbox[j];
        float yy1 = fmaxf(b.x, q.x), xx1 = fmaxf(b.y, q.y);
        float yy2 = fminf(b.z, q.z), xx2 = fminf(b.w, q.w);
        float inter = fmaxf(yy2 - yy1, 0.f) * fmaxf(xx2 - xx1, 0.f);
        // iou > 0.7  <=>  1.7*inter > 0.7*(ar+ac)
        if (1.7f * inter > 0.7f * (ar + carea[j])) bits |= (1u << j);
    }
    mask[(size_t)r * MWORDS + blockIdx.x] = bits;
}

// sequential greedy resolve on ONE wave32: lane L owns mask words L, L+32, ..., L+160
__global__ void nms_seq(const u32* __restrict__ vwords, const u32* __restrict__ mask,
                        const float4* __restrict__ sroi, float4* __restrict__ out) {
    int lane = threadIdx.x;
    u32 rem[6];
    #pragma unroll
    for (int s = 0; s < 6; ++s) {
        int w = lane + 32 * s;
        rem[s] = (w < MWORDS) ? ~vwords[w] : 0xFFFFFFFFu;   // invalid => pre-suppressed
    }
    int cnt = 0;
    for (int i = 0; i < NPRE; ++i) {
        int word = i >> 5, bit = i & 31, owner = word & 31, slot = word >> 5;
        // prefetch a future mask row into caches (global_prefetch_b8)
        int pf = (i + 8 < NPRE) ? (i + 8) : (NPRE - 1);
        if (lane < 6) __builtin_prefetch((const void*)(mask + (size_t)pf * MWORDS + lane * 32), 0, 0);
        u32 myw = rem[0];
        if (slot == 1) myw = rem[1]; else if (slot == 2) myw = rem[2];
        else if (slot == 3) myw = rem[3]; else if (slot == 4) myw = rem[4];
        else if (slot == 5) myw = rem[5];
        u32 rw = __builtin_amdgcn_readlane(myw, owner);     // uniform broadcast, wave32
        if (((rw >> bit) & 1u) == 0u) {                     // box i kept
            if (lane == 0) out[cnt] = sroi[i];
            const u32* row = mask + (size_t)i * MWORDS;
            #pragma unroll
            for (int s = 0; s < 6; ++s) {
                int w = lane + 32 * s;
                u32 m = 0;
                if (w < MWORDS && w >= word) {
                    m = row[w];
                    if (w == word) m &= ~(0xFFFFFFFFu >> (31 - bit));  // only j > i
                }
                rem[s] |= m;
            }
            if (++cnt == NPOST) break;
        }
    }
}

extern "C" void kernel_launch(void* const* d_in, const int* in_sizes, int n_in,
                              void* d_out, int out_size, void* d_ws, size_t ws_size,
                              hipStream_t stream) {
    const float4* loc    = (const float4*)d_in[0];
    const float*  score  = (const float*) d_in[1];
    const float4* anchor = (const float4*)d_in[2];
    const int*    imgh   = (const int*)   d_in[3];
    const int*    imgw   = (const int*)   d_in[4];
    float4*       out    = (float4*)      d_out;

    char* ws = (char*)d_ws;
    float4* roi    = (float4*)(ws + OFF_ROI);
    u32*    key    = (u32*)   (ws + OFF_KEY);
    u32*    hist1  = (u32*)   (ws + OFF_HIST1);
    u32*    hist2  = (u32*)   (ws + OFF_HIST2);
    u32*    hist3  = (u32*)   (ws + OFF_HIST3);
    u32*    sel    = (u32*)   (ws + OFF_SEL);
    u32*    vwords = (u32*)   (ws + OFF_VWORDS);
    u32*    ckey   = (u32*)   (ws + OFF_CKEY);
    u32*    cidx   = (u32*)   (ws + OFF_CIDX);
    float4* sroi   = (float4*)(ws + OFF_SROI);
    u32*    maskb  = (u32*)   (ws + OFF_MASK);

    const int gN = N_ELE / TPB;  // 4096

    // re-init all state every call (determinism under graph replay)
    int zc = (int)((OFF_CKEY - OFF_HIST1) / 4);  // hist1..vwords (incl. pad)
    zero_u32<<<(zc + TPB - 1) / TPB, TPB, 0, stream>>>(hist1, zc);
    zero_u32<<<(NPRE * 4 + TPB - 1) / TPB, TPB, 0, stream>>>((u32*)sroi, NPRE * 4);
    zero_u32<<<(NPOST * 4 + TPB - 1) / TPB, TPB, 0, stream>>>((u32*)out, NPOST * 4);

    decode_hist1<<<gN, TPB, 0, stream>>>(loc, score, anchor, imgh, imgw, roi, key, hist1);
    scan_kernel <<<1, 32, 0, stream>>>(hist1, 4096, sel, 0);
    hist2_kernel<<<gN, TPB, 0, stream>>>(key, sel, hist2);
    scan_kernel <<<1, 32, 0, stream>>>(hist2, 4096, sel, 1);
    hist3_kernel<<<gN, TPB, 0, stream>>>(key, sel, hist3);
    scan_kernel <<<1, 32, 0, stream>>>(hist3, 256, sel, 2);
    gather_kernel<<<gN, TPB, 0, stream>>>(key, sel, ckey, cidx);
    rank_kernel  <<<CAND_CAP / TPB, TPB, 0, stream>>>(ckey, cidx, sel, roi, sroi, vwords);
    nms_mask     <<<dim3(MWORDS, (NPRE + 255) / 256), dim3(32, 8), 0, stream>>>(sroi, maskb);
    nms_seq      <<<1, 32, 0, stream>>>(vwords, maskb, sroi, out);
}